// SVDPlaneProjection_7593502179725
// MI455X (gfx1250) — compile-verified
//
#include <hip/hip_runtime.h>
#include <math.h>

// CDNA5 / gfx1250, wave32. WMMA f32 16x16x4 used for (a) point-plane signed
// distances and (b) masked moment accumulation (M^T * F GEMM).

typedef float v2f __attribute__((ext_vector_type(2)));
typedef float v8f __attribute__((ext_vector_type(8)));

#define THRESH 0.02f

__device__ __forceinline__ v8f zero8() {
  v8f z;
  for (int i = 0; i < 8; i++) z[i] = 0.0f;
  return z;
}

// D = A(16x4) * B(4x16) + C(16x16), all f32.
__device__ __forceinline__ v8f wmma4(v2f a, v2f b, v8f c) {
  return __builtin_amdgcn_wmma_f32_16x16x4_f32(false, a, false, b, (short)0, c,
                                               false, false);
}

// Swap data between lanes l and l^16 (immediate-pattern ds_swizzle SWAPX16:
// xor_mask=0x10, or_mask=0, and_mask=0x1f -> offset 0x401F).
__device__ __forceinline__ float swap16(float x) {
  return __int_as_float(
      __builtin_amdgcn_ds_swizzle(__float_as_int(x), 0x401F));
}

// ---------------------------------------------------------------------------
// Kernel 1: per-plane masked moments via WMMA.
// stats[plane][0..9] = cnt, Sx, Sy, Sz, Sxx, Sxy, Sxz, Syy, Syz, Szz
// ---------------------------------------------------------------------------
__global__ void __launch_bounds__(256) k1_plane_stats(
    const float* __restrict__ pts, const float* __restrict__ nrm,
    const float* __restrict__ dst, float* __restrict__ stats, int nPoints) {
  const int lane = threadIdx.x & 31;
  const int hi = lane >> 4;   // 0 = lanes 0..15, 1 = lanes 16..31
  const int l16 = lane & 15;
  const int waveId = (int)((blockIdx.x * blockDim.x + threadIdx.x) >> 5);
  const int nWaves = (int)((gridDim.x * blockDim.x) >> 5);
  const float hif = (float)hi;  // branchless splice helper

  // B1 tiles: plane data [nx; ny; nz; -d] per 16-plane tile, WMMA B-layout.
  // Branchless: load nrm[3pl + 2*hi] (nx or nz); second element ny or -d.
  v2f b1[4];
#pragma unroll
  for (int t = 0; t < 4; t++) {
    int pl = 16 * t + l16;
    float bx = nrm[3 * pl + 2 * hi];
    float by = nrm[3 * pl + 1];
    float bd = -dst[pl];
    b1[t].x = bx;
    b1[t].y = hi ? bd : by;
  }

  // Moment accumulators: O[plane, feat] as 4 WMMA C/D tiles (planes x 16
  // feats).
  v8f acc[4];
#pragma unroll
  for (int t = 0; t < 4; t++) acc[t] = zero8();

  // Lane-constant feature selectors for feature column f = l16:
  // feat f = (sa . [x,y,z,1]) * (sb . [x,y,z,1])
  //   features: {1, x, y, z, xx, xy, xz, yy, yz, zz, 0...}
  const int f = l16;
  const float sax = (f == 1 || f == 4 || f == 5 || f == 6) ? 1.0f : 0.0f;
  const float say = (f == 2 || f == 7 || f == 8) ? 1.0f : 0.0f;
  const float saz = (f == 3 || f == 9) ? 1.0f : 0.0f;
  const float sa1 = (f == 0) ? 1.0f : 0.0f;
  const float sbx = (f == 4) ? 1.0f : 0.0f;
  const float sby = (f == 5 || f == 7) ? 1.0f : 0.0f;
  const float sbz = (f == 6 || f == 8 || f == 9) ? 1.0f : 0.0f;
  const float sb1 = (f <= 3) ? 1.0f : 0.0f;

  const int nTiles = nPoints >> 4;  // 16 points per wave-tile (4M % 16 == 0)

  for (int tile = waveId; tile < nTiles; tile += nWaves) {
    const int base = tile << 4;
    // A1 tile: [x,y,z,1] for 16 points, WMMA A-layout (k = v + 2*hi).
    // Branchless loads: both addresses always in-bounds.
    const int p = base + l16;
    float ax = pts[3 * p + 2 * hi];  // x (lo) or z (hi)
    float ay = pts[3 * p + 1];       // y (always valid)
    v2f a1;
    a1.x = ax;
    a1.y = hi ? 1.0f : ay;
    __builtin_prefetch(pts + 3 * (base + (nWaves << 4)), 0, 1);

    // dots[pt, pl] = x*nx + y*ny + z*nz - d ; mask = |dot| < T
    v8f m[4];
#pragma unroll
    for (int t = 0; t < 4; t++) {
      v8f d = wmma4(a1, b1[t], zero8());
#pragma unroll
      for (int j = 0; j < 8; j++)
        m[t][j] = (fabsf(d[j]) < THRESH) ? 1.0f : 0.0f;
    }

    // Moment GEMM: O += M^T(16 planes x 4 pts) * F(4 pts x 16 feats),
    // 4 point-chunks of K=4.
#pragma unroll
    for (int c = 0; c < 4; c++) {
      // B2 tile: feature values of points ptB = 4c + v + 2*hi, column f.
      const int pt0 = 4 * c + 2 * hi;
      const int pt1 = pt0 + 1;
      float x0 = __shfl(a1.x, pt0, 32), y0 = __shfl(a1.y, pt0, 32),
            z0 = __shfl(a1.x, pt0 + 16, 32);
      float x1 = __shfl(a1.x, pt1, 32), y1 = __shfl(a1.y, pt1, 32),
            z1 = __shfl(a1.x, pt1 + 16, 32);
      v2f b2;
      b2.x = (sa1 + sax * x0 + say * y0 + saz * z0) *
             (sb1 + sbx * x0 + sby * y0 + sbz * z0);
      b2.y = (sa1 + sax * x1 + say * y1 + saz * z1) *
             (sb1 + sbx * x1 + sby * y1 + sbz * z1);

      const int basep = 4 * (c & 1);
      const bool cLow = (c < 2);
#pragma unroll
      for (int t = 0; t < 4; t++) {
        // Transpose mask D-layout -> A-layout: one half-wave swap / element.
        v2f a2;
#pragma unroll
        for (int v = 0; v < 2; v++) {
          float own = cLow ? m[t][basep + v]     : m[t][basep + v + 2];
          float shs = cLow ? m[t][basep + v + 2] : m[t][basep + v];
          float sh = swap16(shs);
          bool useSh = cLow ? (hi != 0) : (hi == 0);
          a2[v] = useSh ? sh : own;
        }
        acc[t] = wmma4(a2, b2, acc[t]);
      }
    }
  }
  (void)hif;

  // Reduce across waves. D-layout: plane = 16t + j + 8*hi, feat = l16.
#pragma unroll
  for (int t = 0; t < 4; t++) {
#pragma unroll
    for (int j = 0; j < 8; j++) {
      int plane = 16 * t + j + 8 * hi;
      atomicAdd(&stats[plane * 16 + l16], acc[t][j]);
    }
  }
}

// ---------------------------------------------------------------------------
// Kernel 2: 64 independent closed-form symmetric 3x3 eigensolves.
// Stores (valid*rn.xyz, valid*rd) so invalid planes no-op in projection.
// ---------------------------------------------------------------------------
__global__ void k2_plane_solve(const float* __restrict__ stats,
                               const float* __restrict__ nrm,
                               const float* __restrict__ dst,
                               float* __restrict__ planeB) {
  int pl = threadIdx.x;
  if (pl >= 64) return;
  (void)dst;
  const float* s = stats + pl * 16;
  float cnt = s[0];
  float inv = 1.0f / fmaxf(cnt, 1.0f);
  float cx = s[1] * inv, cy = s[2] * inv, cz = s[3] * inv;
  float xx = s[4] - cnt * cx * cx, sxy = s[5] - cnt * cx * cy,
        sxz = s[6] - cnt * cx * cz;
  float yy = s[7] - cnt * cy * cy, syz = s[8] - cnt * cy * cz,
        zz = s[9] - cnt * cz * cz;
  float valid = (cnt >= 3.0f) ? 1.0f : 0.0f;
  float jit = 1.0f - valid;
  xx += jit * 1.0f; yy += jit * 2.0f; zz += jit * 3.0f;

  float n0x = nrm[3 * pl], n0y = nrm[3 * pl + 1], n0z = nrm[3 * pl + 2];
  float rnx = n0x, rny = n0y, rnz = n0z;  // degenerate fallback

  float p1 = sxy * sxy + sxz * sxz + syz * syz;
  float q = (xx + yy + zz) * (1.0f / 3.0f);
  float p2 = (xx - q) * (xx - q) + (yy - q) * (yy - q) + (zz - q) * (zz - q) +
             2.0f * p1;
  if (p2 > 1e-18f) {
    float pp = sqrtf(p2 * (1.0f / 6.0f));
    float ip = 1.0f / pp;
    float b00 = (xx - q) * ip, b11 = (yy - q) * ip, b22 = (zz - q) * ip;
    float b01 = sxy * ip, b02 = sxz * ip, b12 = syz * ip;
    float detB = b00 * (b11 * b22 - b12 * b12) -
                 b01 * (b01 * b22 - b12 * b02) +
                 b02 * (b01 * b12 - b11 * b02);
    float r = fminf(1.0f, fmaxf(-1.0f, 0.5f * detB));
    float phi = acosf(r) * (1.0f / 3.0f);
    float lam = q + 2.0f * pp * cosf(phi + 2.0943951023931953f);  // smallest
    float r0x = xx - lam, r0y = sxy, r0z = sxz;
    float r1x = sxy, r1y = yy - lam, r1z = syz;
    float r2x = sxz, r2y = syz, r2z = zz - lam;
    float c0x = r0y * r1z - r0z * r1y, c0y = r0z * r1x - r0x * r1z,
          c0z = r0x * r1y - r0y * r1x;
    float c1x = r0y * r2z - r0z * r2y, c1y = r0z * r2x - r0x * r2z,
          c1z = r0x * r2y - r0y * r2x;
    float c2x = r1y * r2z - r1z * r2y, c2y = r1z * r2x - r1x * r2z,
          c2z = r1x * r2y - r1y * r2x;
    float nn0 = c0x * c0x + c0y * c0y + c0z * c0z;
    float nn1 = c1x * c1x + c1y * c1y + c1z * c1z;
    float nn2 = c2x * c2x + c2y * c2y + c2z * c2z;
    float bx = c0x, by = c0y, bz = c0z, bn = nn0;
    if (nn1 > bn) { bx = c1x; by = c1y; bz = c1z; bn = nn1; }
    if (nn2 > bn) { bx = c2x; by = c2y; bz = c2z; bn = nn2; }
    if (bn > 1e-24f) {
      float is = rsqrtf(bn);
      rnx = bx * is; rny = by * is; rnz = bz * is;
    }
  }
  float dp = rnx * n0x + rny * n0y + rnz * n0z;
  if (dp < 0.0f) { rnx = -rnx; rny = -rny; rnz = -rnz; }
  float rd = cx * rnx + cy * rny + cz * rnz;
  planeB[4 * pl + 0] = valid * rnx;
  planeB[4 * pl + 1] = valid * rny;
  planeB[4 * pl + 2] = valid * rnz;
  planeB[4 * pl + 3] = valid * rd;
}

// ---------------------------------------------------------------------------
// Kernel 3: masks via WMMA + ballot, then sequential 64-plane projection.
// ---------------------------------------------------------------------------
__global__ void __launch_bounds__(256) k3_project(
    const float* __restrict__ pts, const float* __restrict__ nrm,
    const float* __restrict__ dst, const float4* __restrict__ planeB,
    float* __restrict__ out, int nPoints) {
  const int lane = threadIdx.x & 31;
  const int hi = lane >> 4;
  const int l16 = lane & 15;
  const int waveId = (int)((blockIdx.x * blockDim.x + threadIdx.x) >> 5);
  const int nWaves = (int)((gridDim.x * blockDim.x) >> 5);

  v2f b1[4];
#pragma unroll
  for (int t = 0; t < 4; t++) {
    int pl = 16 * t + l16;
    float bx = nrm[3 * pl + 2 * hi];
    float by = nrm[3 * pl + 1];
    float bd = -dst[pl];
    b1[t].x = bx;
    b1[t].y = hi ? bd : by;
  }

  const int nIter = nPoints >> 5;  // 32 points per wave iter (4M % 32 == 0)
  for (int it = waveId; it < nIter; it += nWaves) {
    const int base = it << 5;
    const int myp = base + lane;  // this lane's projected point
    float qx = pts[3 * myp + 0], qy = pts[3 * myp + 1], qz = pts[3 * myp + 2];
    __builtin_prefetch(pts + 3 * (base + (nWaves << 5)), 0, 1);

    unsigned wm[4] = {0u, 0u, 0u, 0u};
#pragma unroll
    for (int h = 0; h < 2; h++) {
      const int p = base + (h << 4) + l16;
      float ax = pts[3 * p + 2 * hi];  // x or z, branchless
      float ay = pts[3 * p + 1];
      v2f a1;
      a1.x = ax;
      a1.y = hi ? 1.0f : ay;
#pragma unroll
      for (int t = 0; t < 4; t++) {
        v8f d = wmma4(a1, b1[t], zero8());
#pragma unroll
        for (int v = 0; v < 8; v++) {
          unsigned b = __builtin_amdgcn_ballot_w32(fabsf(d[v]) < THRESH);
          bool cap = (hi == h) && ((l16 & 7) == v);
          wm[t] = cap ? b : wm[t];
        }
      }
    }
    const unsigned shct = (l16 & 8) ? 16u : 0u;

#pragma unroll
    for (int t = 0; t < 4; t++) {
      unsigned w = wm[t] >> shct;  // bits 0..15 = planes 16t..16t+15
#pragma unroll
      for (int i = 0; i < 16; i++) {
        const int pl = (t << 4) + i;  // uniform -> s_load_b128 from K$
        float4 r = planeB[pl];
        float eff = (float)((w >> i) & 1u);
        float sdel = eff * (qx * r.x + qy * r.y + qz * r.z - r.w);
        qx -= sdel * r.x; qy -= sdel * r.y; qz -= sdel * r.z;
      }
    }
    out[3 * myp + 0] = qx;
    out[3 * myp + 1] = qy;
    out[3 * myp + 2] = qz;
  }
}

// ---------------------------------------------------------------------------
extern "C" void kernel_launch(void* const* d_in, const int* in_sizes, int n_in,
                              void* d_out, int out_size, void* d_ws,
                              size_t ws_size, hipStream_t stream) {
  (void)n_in; (void)out_size; (void)ws_size;
  const float* pts = (const float*)d_in[0];
  const float* nrm = (const float*)d_in[1];
  const float* dst = (const float*)d_in[2];
  float* out = (float*)d_out;
  const int nPoints = in_sizes[0] / 3;

  float* stats = (float*)d_ws;           // 64 * 16 floats
  float* planeB = stats + 64 * 16;       // 64 * 4 floats (16B aligned)

  hipMemsetAsync(stats, 0, 64 * 16 * sizeof(float), stream);
  k1_plane_stats<<<640, 256, 0, stream>>>(pts, nrm, dst, stats, nPoints);
  k2_plane_solve<<<1, 64, 0, stream>>>(stats, nrm, dst, planeB);
  k3_project<<<1024, 256, 0, stream>>>(pts, nrm, dst, (const float4*)planeB,
                                       out, nPoints);
}